// GCN_Encoder_17849884082524
// MI455X (gfx1250) — compile-verified
//
#include <hip/hip_runtime.h>
#include <math.h>

typedef __attribute__((ext_vector_type(2))) float v2f;
typedef __attribute__((ext_vector_type(8))) float v8f;

// ---------------------------------------------------------------------------
// Degree / normalization kernels
// ---------------------------------------------------------------------------
__global__ void zero_int_kernel(int* __restrict__ p, int n) {
    int i = blockIdx.x * blockDim.x + threadIdx.x;
    if (i < n) p[i] = 0;
}

__global__ void deg_count_kernel(const int* __restrict__ dst, int* __restrict__ deg, int nEdges) {
    int i = blockIdx.x * blockDim.x + threadIdx.x;
    if (i < nEdges) atomicAdd(&deg[dst[i]], 1);
}

__global__ void make_dinv_kernel(const int* __restrict__ deg, float* __restrict__ dinv, int n) {
    int i = blockIdx.x * blockDim.x + threadIdx.x;
    if (i < n) {
        // +1 for the self loop; degree is therefore always > 0
        dinv[i] = rsqrtf((float)(deg[i] + 1));
    }
}

// ---------------------------------------------------------------------------
// Register-blocked f32 WMMA GEMM:  C[M,N] = A[M,K] @ B[K,N], row-major.
// N == NT*16 exactly; M % 16 == 0; K % 4 == 0.
// One wave32 computes a full 16 x N row-panel: NT independent 16x16
// accumulators (good XDL pipelining), A fragment loaded once per K-step
// (8x less A traffic than tile-per-wave). Exit guard is wave-uniform so
// EXEC is all-ones at every V_WMMA (ISA 7.12 requirement).
// ---------------------------------------------------------------------------
template <int NT>
__global__ void __launch_bounds__(128)
gemm_wmma_f32_rowpanel(const float* __restrict__ A, const float* __restrict__ B,
                       float* __restrict__ C, int K, int tilesM) {
    const int lane   = threadIdx.x & 31;
    const int waveId = blockIdx.x * (blockDim.x >> 5) + (threadIdx.x >> 5);
    if (waveId >= tilesM) return;          // uniform across the wave

    const int N     = NT * 16;
    const int mRow  = waveId * 16 + (lane & 15);  // A row for this lane
    const int nLane = lane & 15;                  // column-within-tile
    const int kHalf = (lane >> 4) * 2;            // lanes 0-15 -> K+0/1, 16-31 -> K+2/3

    const float* __restrict__ Arow = A + (size_t)mRow * K;

    v8f acc[NT];
#pragma unroll
    for (int t = 0; t < NT; ++t) acc[t] = (v8f){};

    for (int k0 = 0; k0 < K; k0 += 4) {
        // A-matrix 16x4 layout (ISA 7.12.2): VGPR0 = K=0|2, VGPR1 = K=1|3
        v2f a;
        a.x = Arow[k0 + kHalf];
        a.y = Arow[k0 + kHalf + 1];

        const float* __restrict__ Brow0 = B + (size_t)(k0 + kHalf) * N;
        const float* __restrict__ Brow1 = Brow0 + N;
#pragma unroll
        for (int t = 0; t < NT; ++t) {
            v2f b;
            b.x = Brow0[t * 16 + nLane];
            b.y = Brow1[t * 16 + nLane];
            acc[t] = __builtin_amdgcn_wmma_f32_16x16x4_f32(
                /*neg_a=*/false, a, /*neg_b=*/false, b,
                /*c_mod=*/(short)0, acc[t], /*reuse_a=*/false, /*reuse_b=*/false);
        }
    }

    // C/D layout: VGPR r -> row r (lanes 0-15) / row r+8 (lanes 16-31)
    const int mBase = waveId * 16 + (lane >> 4) * 8;
#pragma unroll
    for (int t = 0; t < NT; ++t) {
        const int nCol = t * 16 + nLane;
#pragma unroll
        for (int r = 0; r < 8; ++r) {
            C[(size_t)(mBase + r) * N + nCol] = acc[t][r];
        }
    }
}

// ---------------------------------------------------------------------------
// out[i,:] = h[i,:] * dinv[i]^2 + bias      (self-loop contribution + bias;
// also deterministically initializes every element of the output buffer)
// ---------------------------------------------------------------------------
__global__ void self_loop_bias_kernel(const float* __restrict__ h,
                                      const float* __restrict__ dinv,
                                      const float* __restrict__ bias,
                                      float* __restrict__ out,
                                      int nNodes, int C4) {
    int tid  = blockIdx.x * blockDim.x + threadIdx.x;
    int node = tid / C4;
    if (node >= nNodes) return;
    int c4 = tid - node * C4;

    float di = dinv[node];
    float w  = di * di;
    float4 hv = ((const float4*)h)[(size_t)tid];
    float4 bv = ((const float4*)bias)[c4];
    float4 o;
    o.x = hv.x * w + bv.x;
    o.y = hv.y * w + bv.y;
    o.z = hv.z * w + bv.z;
    o.w = hv.w * w + bv.w;
    ((float4*)out)[(size_t)tid] = o;
}

// ---------------------------------------------------------------------------
// Edge-parallel scatter-add:  out[dst,:] += h[src,:] * dinv[src]*dinv[dst]
// One thread per (edge, 4 channels): coalesced 512B row gathers, f32 atomics
// land in the 192MB L2 (out buffers are 51MB / 26MB -> on-die RMW).
// ---------------------------------------------------------------------------
__global__ void edge_aggregate_kernel(const float* __restrict__ h,
                                      const int* __restrict__ src,
                                      const int* __restrict__ dst,
                                      const float* __restrict__ dinv,
                                      float* __restrict__ out,
                                      int nEdges, int C4) {
    int tid = blockIdx.x * blockDim.x + threadIdx.x;
    int e   = tid / C4;
    if (e >= nEdges) return;
    int c4 = tid - e * C4;

    int s = src[e];
    int d = dst[e];
    float norm = dinv[s] * dinv[d];

    float4 hv = ((const float4*)h)[(size_t)s * C4 + c4];
    float* op = out + ((size_t)d * C4 + c4) * 4;
    atomicAdd(op + 0, hv.x * norm);
    atomicAdd(op + 1, hv.y * norm);
    atomicAdd(op + 2, hv.z * norm);
    atomicAdd(op + 3, hv.w * norm);
}

// ---------------------------------------------------------------------------
// In-place tanh (vectorized)
// ---------------------------------------------------------------------------
__global__ void tanh_inplace_kernel(float* __restrict__ p, int n4) {
    int i = blockIdx.x * blockDim.x + threadIdx.x;
    if (i >= n4) return;
    float4 v = ((float4*)p)[(size_t)i];
    v.x = tanhf(v.x);
    v.y = tanhf(v.y);
    v.z = tanhf(v.z);
    v.w = tanhf(v.w);
    ((float4*)p)[(size_t)i] = v;
}

// ---------------------------------------------------------------------------
// Launcher
// ---------------------------------------------------------------------------
extern "C" void kernel_launch(void* const* d_in, const int* in_sizes, int n_in,
                              void* d_out, int out_size, void* d_ws, size_t ws_size,
                              hipStream_t stream) {
    const float* x   = (const float*)d_in[0];
    const int*   ei  = (const int*)d_in[1];
    const float* W1  = (const float*)d_in[2];
    const float* b1  = (const float*)d_in[3];
    const float* W2  = (const float*)d_in[4];
    const float* b2  = (const float*)d_in[5];
    float*       out = (float*)d_out;

    const int HID = in_sizes[3];              // 128
    const int OUT = in_sizes[5];              // 64
    const int IN  = in_sizes[2] / HID;        // 256
    const int Nn  = in_sizes[0] / IN;         // 100000
    const int E   = in_sizes[1] / 2;          // 1600000

    const int* src = ei;       // edge_index[0]
    const int* dst = ei + E;   // edge_index[1]

    // --- carve workspace ---------------------------------------------------
    char*  ws  = (char*)d_ws;
    size_t off = 0;
    auto carve = [&](size_t bytes) -> char* {
        char* p = ws + off;
        off = (off + bytes + 255) & ~(size_t)255;
        return p;
    };
    int*   deg  = (int*)  carve((size_t)Nn * sizeof(int));
    float* dinv = (float*)carve((size_t)Nn * sizeof(float));
    float* h1   = (float*)carve((size_t)Nn * HID * sizeof(float));
    float* out1 = (float*)carve((size_t)Nn * HID * sizeof(float));
    float* h2   = (float*)carve((size_t)Nn * OUT * sizeof(float));
    (void)ws_size;

    const int T = 256;

    // --- degree + symmetric normalization (shared by both layers) ----------
    zero_int_kernel<<<(Nn + T - 1) / T, T, 0, stream>>>(deg, Nn);
    deg_count_kernel<<<(E + T - 1) / T, T, 0, stream>>>(dst, deg, E);
    make_dinv_kernel<<<(Nn + T - 1) / T, T, 0, stream>>>(deg, dinv, Nn);

    const int tilesM = Nn / 16;               // 6250 (Nn % 16 == 0)
    const int gemmThreads = 128;              // 4 waves / block
    const int gemmBlocks  = (tilesM + 3) / 4;

    // --- layer 1: h1 = x @ W1 ; out1 = propagate(h1) + b1 ; tanh -----------
    {
        gemm_wmma_f32_rowpanel<8><<<gemmBlocks, gemmThreads, 0, stream>>>(
            x, W1, h1, IN, tilesM);           // N = 128 = 8*16

        int c4 = HID / 4;
        int totInit = Nn * c4;
        self_loop_bias_kernel<<<(totInit + T - 1) / T, T, 0, stream>>>(h1, dinv, b1, out1, Nn, c4);

        long long totEdge = (long long)E * c4;
        edge_aggregate_kernel<<<(int)((totEdge + T - 1) / T), T, 0, stream>>>(
            h1, src, dst, dinv, out1, E, c4);

        tanh_inplace_kernel<<<(totInit + T - 1) / T, T, 0, stream>>>(out1, totInit);
    }

    // --- layer 2: h2 = out1 @ W2 ; d_out = propagate(h2) + b2 --------------
    {
        gemm_wmma_f32_rowpanel<4><<<gemmBlocks, gemmThreads, 0, stream>>>(
            out1, W2, h2, HID, tilesM);       // N = 64 = 4*16

        int c4 = OUT / 4;
        int totInit = Nn * c4;
        self_loop_bias_kernel<<<(totInit + T - 1) / T, T, 0, stream>>>(h2, dinv, b2, out, Nn, c4);

        long long totEdge = (long long)E * c4;
        edge_aggregate_kernel<<<(int)((totEdge + T - 1) / T), T, 0, stream>>>(
            h2, src, dst, dinv, out, E, c4);
    }
}